// ESN_14843406975749
// MI455X (gfx1250) — compile-verified
//
#include <hip/hip_runtime.h>
#include <math.h>

// Problem constants (fixed by the reference).
#define SEQL   1024
#define I_DIM  256
#define H_DIM  4096
#define NPRED  256
#define EXT_K  (1 + I_DIM + H_DIM)   // 4353

typedef float v2f __attribute__((ext_vector_type(2)));
typedef float v8f __attribute__((ext_vector_type(8)));

// ---------------------------------------------------------------------------
// Kernel 1: U[t][h] = in_bias[h] + sum_i inputs[t][i] * in_weight[h][i]
// GEMM M=1024 (t), N=4096 (h), K=256 via V_WMMA_F32_16X16X4_F32.
// One wave per 16x16 tile; 8 waves (tiles) per 256-thread block.
// ---------------------------------------------------------------------------
__global__ __launch_bounds__(256)
void esn_inproj_wmma(const float* __restrict__ inputs,
                     const float* __restrict__ in_weight,
                     const float* __restrict__ in_bias,
                     float* __restrict__ U) {
  const int wave = threadIdx.x >> 5;
  const int lane = threadIdx.x & 31;
  const int tile = blockIdx.x * 8 + wave;
  const int nTilesH = H_DIM / 16;
  const int t0 = (tile / nTilesH) * 16;
  const int h0 = (tile % nTilesH) * 16;

  const int mn   = lane & 15;      // M index for A, N index for B/D
  const int half = lane >> 4;      // half-wave selects K pair (A/B), M+8 (D)
  const int kb   = half * 2;

  v8f c;
  const float bias = in_bias[h0 + mn];
#pragma unroll
  for (int r = 0; r < 8; ++r) c[r] = bias;

  const float* arow = inputs    + (size_t)(t0 + mn) * I_DIM;
  const float* brow = in_weight + (size_t)(h0 + mn) * I_DIM;

  for (int k = 0; k < I_DIM; k += 4) {
    v2f a = *(const v2f*)(arow + k + kb);   // A[m][k+kb], A[m][k+kb+1]
    v2f b = *(const v2f*)(brow + k + kb);   // B[k+kb][n], B[k+kb+1][n] (= W_in^T)
    c = __builtin_amdgcn_wmma_f32_16x16x4_f32(false, a, false, b,
                                              (short)0, c, false, false);
  }

#pragma unroll
  for (int r = 0; r < 8; ++r) {
    const int mm = r + 8 * half;
    U[(size_t)(t0 + mm) * H_DIM + h0 + mn] = c[r];
  }
}

// ---------------------------------------------------------------------------
// Kernel 2: teacher outputs for all 1024 steps:
//   out[t][o] = w_out[o][0] + inputs[t]·w_out[o][1:257] + states[t]·w_out[o][257:]
// GEMM M=1024, N=256, K=4353 via WMMA.
// ---------------------------------------------------------------------------
__global__ __launch_bounds__(256)
void esn_outproj_wmma(const float* __restrict__ inputs,
                      const float* __restrict__ states,
                      const float* __restrict__ w_out,
                      float* __restrict__ out) {
  const int wave = threadIdx.x >> 5;
  const int lane = threadIdx.x & 31;
  const int tile = blockIdx.x * 8 + wave;
  const int nTilesO = I_DIM / 16;
  const int t0 = (tile / nTilesO) * 16;
  const int o0 = (tile % nTilesO) * 16;

  const int mn   = lane & 15;
  const int half = lane >> 4;
  const int kb   = half * 2;

  const float* wrow = w_out + (size_t)(o0 + mn) * EXT_K;

  v8f c;
  const float bias = wrow[0];             // the "ones" column of ext
#pragma unroll
  for (int r = 0; r < 8; ++r) c[r] = bias;

  // K part 1: inputs (K=256), w_out columns 1..256
  {
    const float* arow = inputs + (size_t)(t0 + mn) * I_DIM;
    for (int k = 0; k < I_DIM; k += 4) {
      v2f a = *(const v2f*)(arow + k + kb);
      v2f b;
      b[0] = wrow[1 + k + kb];
      b[1] = wrow[2 + k + kb];
      c = __builtin_amdgcn_wmma_f32_16x16x4_f32(false, a, false, b,
                                                (short)0, c, false, false);
    }
  }
  // K part 2: states (K=4096), w_out columns 257..4352
  {
    const float* arow = states + (size_t)(t0 + mn) * H_DIM;
    for (int k = 0; k < H_DIM; k += 4) {
      v2f a = *(const v2f*)(arow + k + kb);
      v2f b;
      b[0] = wrow[1 + I_DIM + k + kb];
      b[1] = wrow[2 + I_DIM + k + kb];
      c = __builtin_amdgcn_wmma_f32_16x16x4_f32(false, a, false, b,
                                                (short)0, c, false, false);
    }
  }

#pragma unroll
  for (int r = 0; r < 8; ++r) {
    const int mm = r + 8 * half;
    out[(size_t)(t0 + mm) * I_DIM + o0 + mn] = c[r];
  }
}

// ---------------------------------------------------------------------------
// Kernel 3: one recurrence step (the sequential bottleneck).
//   state_out = tanh(base + W_res · state_in)
//   base = U[t] (teacher, precomputed) or in_bias + W_in·inp (prediction).
// Bandwidth-bound matvec: state staged once in LDS, one wave per output row,
// 256 threads = 8 rows per block, coalesced float4 streams of W_res from L2.
// ---------------------------------------------------------------------------
__global__ __launch_bounds__(256)
void esn_step(const float* __restrict__ res_weight,
              const float* __restrict__ state_in,
              float* __restrict__ state_out,
              const float* __restrict__ u_row,      // teacher mode (else null)
              const float* __restrict__ inp,        // prediction mode (else null)
              const float* __restrict__ in_weight,
              const float* __restrict__ in_bias,
              float* __restrict__ state_log) {      // optional state logging
  __shared__ __align__(16) float s_state[H_DIM];
  __shared__ __align__(16) float s_inp[I_DIM];

  for (int i = threadIdx.x * 4; i < H_DIM; i += blockDim.x * 4)
    *(float4*)(s_state + i) = *(const float4*)(state_in + i);
  if (inp)
    for (int i = threadIdx.x; i < I_DIM; i += blockDim.x) s_inp[i] = inp[i];
  __syncthreads();

  const int wave = threadIdx.x >> 5;
  const int lane = threadIdx.x & 31;
  const int row  = blockIdx.x * 8 + wave;

  const float* wrow = res_weight + (size_t)row * H_DIM;
  float sum = 0.0f;
  for (int k = lane * 4; k < H_DIM; k += 32 * 4) {
    float4 w = *(const float4*)(wrow + k);
    float4 s = *(const float4*)(s_state + k);
    sum = fmaf(w.x, s.x, sum);
    sum = fmaf(w.y, s.y, sum);
    sum = fmaf(w.z, s.z, sum);
    sum = fmaf(w.w, s.w, sum);
  }

  float base;
  if (inp) {
    const float* irow = in_weight + (size_t)row * I_DIM;
    for (int k = lane * 4; k < I_DIM; k += 32 * 4) {
      float4 w = *(const float4*)(irow + k);
      float4 s = *(const float4*)(s_inp + k);
      sum = fmaf(w.x, s.x, sum);
      sum = fmaf(w.y, s.y, sum);
      sum = fmaf(w.z, s.z, sum);
      sum = fmaf(w.w, s.w, sum);
    }
    base = in_bias[row];
  } else {
    base = u_row[row];   // already includes in_bias
  }

#pragma unroll
  for (int off = 16; off > 0; off >>= 1)
    sum += __shfl_xor(sum, off, 32);

  if (lane == 0) {
    const float v = tanhf(base + sum);
    state_out[row] = v;
    if (state_log) state_log[row] = v;
  }
}

// ---------------------------------------------------------------------------
// Kernel 4: prediction-phase output matvec: out = W_out · [1, inp, state]
// One wave per output (256 outputs); state + inp staged in LDS.
// ---------------------------------------------------------------------------
__global__ __launch_bounds__(256)
void esn_out(const float* __restrict__ inp,
             const float* __restrict__ state,
             const float* __restrict__ w_out,
             float* __restrict__ out) {
  __shared__ __align__(16) float s_state[H_DIM];
  __shared__ __align__(16) float s_inp[I_DIM];

  for (int i = threadIdx.x * 4; i < H_DIM; i += blockDim.x * 4)
    *(float4*)(s_state + i) = *(const float4*)(state + i);
  for (int i = threadIdx.x; i < I_DIM; i += blockDim.x) s_inp[i] = inp[i];
  __syncthreads();

  const int wave = threadIdx.x >> 5;
  const int lane = threadIdx.x & 31;
  const int row  = blockIdx.x * 8 + wave;

  const float* wrow = w_out + (size_t)row * EXT_K;
  float sum = (lane == 0) ? wrow[0] : 0.0f;
  for (int k = lane; k < I_DIM; k += 32)
    sum = fmaf(s_inp[k], wrow[1 + k], sum);
  for (int k = lane; k < H_DIM; k += 32)
    sum = fmaf(s_state[k], wrow[1 + I_DIM + k], sum);

#pragma unroll
  for (int off = 16; off > 0; off >>= 1)
    sum += __shfl_xor(sum, off, 32);

  if (lane == 0) out[row] = sum;
}

// ---------------------------------------------------------------------------
extern "C" void kernel_launch(void* const* d_in, const int* in_sizes, int n_in,
                              void* d_out, int out_size, void* d_ws, size_t ws_size,
                              hipStream_t stream) {
  (void)in_sizes; (void)n_in; (void)out_size; (void)ws_size;

  const float* inputs     = (const float*)d_in[0];  // [1024,1,256]
  const float* state0     = (const float*)d_in[1];  // [1,4096] zeros
  const float* in_weight  = (const float*)d_in[2];  // [4096,256]
  const float* res_weight = (const float*)d_in[3];  // [4096,4096]
  const float* in_bias    = (const float*)d_in[4];  // [4096]
  const float* w_out      = (const float*)d_in[5];  // [256,4353]
  float*       out        = (float*)d_out;          // [1280,1,256]

  float* ws     = (float*)d_ws;
  float* U      = ws;                                   // SEQL*H_DIM
  float* states = U + (size_t)SEQL * H_DIM;             // SEQL*H_DIM
  float* stA    = states + (size_t)SEQL * H_DIM;        // H_DIM
  float* stB    = stA + H_DIM;                          // H_DIM

  // Initial state (zeros) from input — async d2d only (graph-capture safe).
  hipMemcpyAsync(stA, state0, H_DIM * sizeof(float),
                 hipMemcpyDeviceToDevice, stream);

  // Phase 1: batched input projection (WMMA GEMM), lifted out of the loop.
  {
    const int tiles = (SEQL / 16) * (H_DIM / 16);       // 16384
    esn_inproj_wmma<<<tiles / 8, 256, 0, stream>>>(inputs, in_weight, in_bias, U);
  }

  // Phase 2: sequential teacher-forced recurrence (1024 dependent matvecs).
  float* cur = stA;
  float* nxt = stB;
  for (int t = 0; t < SEQL; ++t) {
    esn_step<<<H_DIM / 8, 256, 0, stream>>>(
        res_weight, cur, nxt,
        U + (size_t)t * H_DIM, nullptr, in_weight, in_bias,
        states + (size_t)t * H_DIM);
    float* tmp = cur; cur = nxt; nxt = tmp;
  }

  // Phase 3: batched teacher outputs (WMMA GEMM) -> d_out rows [0,1024).
  {
    const int tiles = (SEQL / 16) * (I_DIM / 16);       // 1024
    esn_outproj_wmma<<<tiles / 8, 256, 0, stream>>>(inputs, states, w_out, out);
  }

  // Phase 4: autoregressive prediction, feeding outputs back as inputs.
  for (int p = 0; p < NPRED; ++p) {
    const float* pin = out + (size_t)(SEQL - 1 + p) * I_DIM;
    esn_step<<<H_DIM / 8, 256, 0, stream>>>(
        res_weight, cur, nxt,
        nullptr, pin, in_weight, in_bias, nullptr);
    esn_out<<<I_DIM / 8, 256, 0, stream>>>(
        pin, nxt, w_out, out + (size_t)(SEQL + p) * I_DIM);
    float* tmp = cur; cur = nxt; nxt = tmp;
  }
}